// MoEGate_76845554860060
// MI455X (gfx1250) — compile-verified
//
#include <hip/hip_runtime.h>
#include <hip/hip_bf16.h>

typedef __attribute__((ext_vector_type(2))) float v2f;
typedef __attribute__((ext_vector_type(8))) float v8f;

#define H      2048
#define NEXP   64
#define TOPK   6
#define SEQLEN 4096
#define ALPHA  1e-3f

// ---------------------------------------------------------------------------
// Zero the aux-loss accumulators (counts[256] + score_sums[256] in d_ws).
// ---------------------------------------------------------------------------
__global__ void moe_zero_kernel(float* acc) {
    int i = threadIdx.x;                 // 512 threads
    if (i < 512) acc[i] = 0.0f;
}

// ---------------------------------------------------------------------------
// Fused router: WMMA fp32 GEMM (16 tok x 64 experts per wave, K=2048) +
// softmax + top-6 + renorm + aux-loss partial reductions.
// blockDim = 128 (4 waves), each block = 64 consecutive tokens (one batch).
// ---------------------------------------------------------------------------
__global__ __launch_bounds__(128) void moe_gate_kernel(
    const float* __restrict__ x,      // [T, H]
    const float* __restrict__ w,      // [NEXP, H]
    int*   __restrict__ out_idx,      // [T, TOPK]
    float* __restrict__ out_w,        // [T, TOPK]
    float* __restrict__ g_cnt,        // [B, NEXP]
    float* __restrict__ g_sum)        // [B, NEXP]
{
    __shared__ float s_logits[4][16][65];   // padded stride 65: conflict-free row scans
    __shared__ float s_cnt[NEXP];

    const int tid  = threadIdx.x;
    const int wave = tid >> 5;
    const int lane = tid & 31;
    const int half = lane >> 4;     // 0: K 0..1 / 1: K 2..3 (fp32 A/B VGPR layout)
    const int l16  = lane & 15;

    if (tid < NEXP) s_cnt[tid] = 0.0f;

    // ---- WMMA GEMM: 16 tokens x 64 experts, K = 2048 --------------------
    const int blockTok0 = blockIdx.x * 64;
    const int token0    = blockTok0 + wave * 16;

    const float* arow = x + (size_t)(token0 + l16) * H + half * 2;
    const float* b0r  = w + (size_t)(l16 +  0) * H + half * 2;
    const float* b1r  = w + (size_t)(l16 + 16) * H + half * 2;
    const float* b2r  = w + (size_t)(l16 + 32) * H + half * 2;
    const float* b3r  = w + (size_t)(l16 + 48) * H + half * 2;

    v8f c0 = {}, c1 = {}, c2 = {}, c3 = {};

    for (int k = 0; k < H; k += 4) {
        v2f a  = *(const v2f*)(arow + k);
        v2f b0 = *(const v2f*)(b0r + k);
        v2f b1 = *(const v2f*)(b1r + k);
        v2f b2 = *(const v2f*)(b2r + k);
        v2f b3 = *(const v2f*)(b3r + k);
        c0 = __builtin_amdgcn_wmma_f32_16x16x4_f32(false, a, false, b0, (short)0, c0, false, false);
        c1 = __builtin_amdgcn_wmma_f32_16x16x4_f32(false, a, false, b1, (short)0, c1, false, false);
        c2 = __builtin_amdgcn_wmma_f32_16x16x4_f32(false, a, false, b2, (short)0, c2, false, false);
        c3 = __builtin_amdgcn_wmma_f32_16x16x4_f32(false, a, false, b3, (short)0, c3, false, false);
    }

    // C layout: VGPR j -> lanes 0-15: (M=j, N=l16), lanes 16-31: (M=j+8, N=l16)
    const int mrow = half * 8;
#pragma unroll
    for (int j = 0; j < 8; ++j) {
        s_logits[wave][mrow + j][ 0 + l16] = c0[j];
        s_logits[wave][mrow + j][16 + l16] = c1[j];
        s_logits[wave][mrow + j][32 + l16] = c2[j];
        s_logits[wave][mrow + j][48 + l16] = c3[j];
    }
    __syncthreads();

    // ---- per-token softmax + top-6 + renorm (threads 0..63, 1 token each)
    if (tid < 64) {
        const int t    = blockTok0 + tid;
        float* row = &s_logits[tid >> 4][tid & 15][0];

        float mx = -3.402823466e+38f;
#pragma unroll 8
        for (int e = 0; e < NEXP; ++e) mx = fmaxf(mx, row[e]);

        float sum = 0.0f;
#pragma unroll 8
        for (int e = 0; e < NEXP; ++e) {
            float v = __expf(row[e] - mx);
            row[e] = v;                 // keep exp values (order-preserving)
            sum += v;
        }
        const float inv = 1.0f / sum;

        unsigned long long chosen = 0ull;
        int   ti[TOPK];
        float tw[TOPK];
        float wsum = 0.0f;
#pragma unroll
        for (int i = 0; i < TOPK; ++i) {
            float best = -1.0f;
            int   bi   = 0;
            for (int e = 0; e < NEXP; ++e) {
                if (!((chosen >> e) & 1ull)) {
                    float v = row[e];
                    if (v > best) { best = v; bi = e; }
                }
            }
            chosen |= 1ull << bi;
            float sc = best * inv;      // softmax score of selected expert
            ti[i] = bi;
            tw[i] = sc;
            wsum += sc;
        }
        const float rn = 1.0f / (wsum + 1e-20f);
#pragma unroll
        for (int i = 0; i < TOPK; ++i) {
            out_idx[(size_t)t * TOPK + i] = ti[i];
            out_w  [(size_t)t * TOPK + i] = tw[i] * rn;
            atomicAdd(&s_cnt[ti[i]], 1.0f);
        }

        // scale row to full softmax scores for the column sums below
#pragma unroll 8
        for (int e = 0; e < NEXP; ++e) row[e] *= inv;
    }
    __syncthreads();

    // ---- aux-loss partials: per-block expert column sums + counts -------
    if (tid < NEXP) {
        const int b = blockTok0 / SEQLEN;   // whole block within one batch
        float acc = 0.0f;
        const float* base = &s_logits[0][0][0];
#pragma unroll 8
        for (int j = 0; j < 64; ++j) acc += base[j * 65 + tid];
        atomicAdd(&g_sum[b * NEXP + tid], acc);
        atomicAdd(&g_cnt[b * NEXP + tid], s_cnt[tid]);
    }
}

// ---------------------------------------------------------------------------
// aux_loss = mean_b sum_e (cnt*E/(S*K)) * (score_sum/S) * ALPHA
// ---------------------------------------------------------------------------
__global__ void moe_aux_kernel(const float* __restrict__ g_cnt,
                               const float* __restrict__ g_sum,
                               float* __restrict__ out_aux, int nb) {
    __shared__ float red[256];
    int tid = threadIdx.x;               // 256 = B * NEXP
    float v = 0.0f;
    if (tid < nb * NEXP) {
        float ce = g_cnt[tid] * ((float)NEXP / ((float)SEQLEN * (float)TOPK));
        float ms = g_sum[tid] * (1.0f / (float)SEQLEN);
        v = ce * ms;
    }
    red[tid] = v;
    __syncthreads();
    for (int s = 128; s > 0; s >>= 1) {
        if (tid < s) red[tid] += red[tid + s];
        __syncthreads();
    }
    if (tid == 0) out_aux[0] = red[0] * (1.0f / (float)nb) * ALPHA;
}

// ---------------------------------------------------------------------------
extern "C" void kernel_launch(void* const* d_in, const int* in_sizes, int n_in,
                              void* d_out, int out_size, void* d_ws, size_t ws_size,
                              hipStream_t stream) {
    const float* x = (const float*)d_in[0];   // [B, S, H] fp32
    const float* w = (const float*)d_in[1];   // [NEXP, H] fp32

    const int T  = in_sizes[0] / H;           // 16384 tokens
    const int nb = T / SEQLEN;                // batch = 4

    float* g_cnt = (float*)d_ws;              // [nb*NEXP] (<=256)
    float* g_sum = g_cnt + 256;               // [nb*NEXP]

    int*   out_idx = (int*)d_out;                       // [T, TOPK] int32 bits
    float* out_w   = (float*)d_out + (size_t)T * TOPK;  // [T, TOPK] fp32
    float* out_aux = (float*)d_out + (size_t)2 * T * TOPK;

    moe_zero_kernel<<<1, 512, 0, stream>>>(g_cnt);
    moe_gate_kernel<<<T / 64, 128, 0, stream>>>(x, w, out_idx, out_w, g_cnt, g_sum);
    moe_aux_kernel<<<1, 256, 0, stream>>>(g_cnt, g_sum, out_aux, nb);
}